// ProposalModule_38147899523160
// MI455X (gfx1250) — compile-verified
//
#include <hip/hip_runtime.h>
#include <hip/hip_bf16.h>
#include <math.h>

// ---------------------------------------------------------------------------
// MI455X (gfx1250) implementation. All GEMMs use native fp32 WMMA
// (v_wmma_f32_16x16x4_f32) to match the fp32 reference exactly in precision.
// Problem footprint (~18MB ws) lives entirely in the 192MB L2 -> compute bound.
// wave32: 256-thread blocks = 8 waves; each wave owns a 16-row output stripe.
// Round 3: W panels also staged in LDS (k-pair float2) so the WMMA inner loop
// is pure ds_load_b64 + v_wmma with immediate offsets (no 64-bit address math,
// no EXEC predication). GMM kernel stages G_k in LDS (64KB) the same way.
// ---------------------------------------------------------------------------

#define BATCH 16
#define NPTS  256
#define CCH   128
#define NCLSC 18
#define KNBRC 16
#define EDG   (NPTS*KNBRC)   // 4096
#define KGC_K 25
#define KGC   (KGC_K*CCH)    // 3200
#define NSCC  18
#define COUT3 97
#define KP    32
#define BN_SCALE_C 0.999995000037499375f
#define PI_F  3.14159265358979323846f
#define GEPS_F 1e-15f
#define NEG_BIG -3.402823466e38f

typedef float v2f __attribute__((ext_vector_type(2)));
typedef float v8f __attribute__((ext_vector_type(8)));

// ---------------------------------------------------------------------------
// Generic fp32 WMMA GEMM:  Y[b][o][n] = act( bn( W[o][:] . X[b][:][n] + bias ) )
// W indexed via (wRow,wCol) strides (+ optional batch stride), X via
// (xBatch,xRow,xCol). Output written channel-major with yRowStride.
// One workgroup: 1 batch x 128 output rows x 256 cols. Both operands staged in
// LDS per K-panel, k-pair packed as float2 -> inner loop is ds_load_b64 + wmma.
// ---------------------------------------------------------------------------
__global__ __launch_bounds__(256) void gemm_ws(
    const float* __restrict__ W, long wBatch, int wRow, int wCol,
    const float* __restrict__ X, long xBatch, int xRow, int xCol,
    const float* __restrict__ bias, const float* __restrict__ bnG,
    const float* __restrict__ bnB,
    float* __restrict__ Y, long yBatch, int yRowStride,
    int Co, int Ci, int doRelu)
{
    __shared__ v2f xs2[KP / 2][NPTS];   // 32KB X panel, k-pair interleaved
    __shared__ v2f ws2[128][KP / 2];    // 16KB W panel, k-pair interleaved

    const int b      = blockIdx.x;
    const int coBase = blockIdx.y * 128;
    const int tid    = threadIdx.x;
    const int wave   = tid >> 5;
    const int lane   = tid & 31;
    const int laneHalf = lane >> 4;     // 0 or 1
    const int lane15   = lane & 15;

    const float* Xb = X + (long)b * xBatch;
    const float* Wb = W + (long)b * wBatch;

    const bool waveActive = (coBase + wave * 16) < Co;     // wave-uniform

    v8f acc[16];
    const v8f vzero = {0.f,0.f,0.f,0.f,0.f,0.f,0.f,0.f};
#pragma unroll
    for (int t = 0; t < 16; ++t) acc[t] = vzero;

    const int nPanels = (Ci + KP - 1) / KP;
    for (int p = 0; p < nPanels; ++p) {
        const int kBase = p * KP;
        // --- stage X panel: thread tid owns column n=tid for all 32 rows,
        //     packing k-pairs into float2 (zero-padded past Ci). Prefetch the
        //     next panel's rows into L2 (global_prefetch_b8).
        {
            const int n = tid;
            for (int rp = 0; rp < KP / 2; ++rp) {
                const int k0 = kBase + 2 * rp;
                v2f v;
                v.x = (k0     < Ci) ? Xb[(long)k0       * xRow + (long)n * xCol] : 0.f;
                v.y = (k0 + 1 < Ci) ? Xb[(long)(k0 + 1) * xRow + (long)n * xCol] : 0.f;
                xs2[rp][n] = v;
                if (k0 + KP < Ci)
                    __builtin_prefetch(Xb + (long)(k0 + KP)     * xRow + (long)n * xCol, 0, 1);
                if (k0 + KP + 1 < Ci)
                    __builtin_prefetch(Xb + (long)(k0 + KP + 1) * xRow + (long)n * xCol, 0, 1);
            }
        }
        // --- stage W panel: 128 rows x 16 k-pairs, all Co/Ci guards applied
        //     here so the compute loop needs no clamps at all.
        for (int i = tid; i < 128 * (KP / 2); i += 256) {
            const int r   = i >> 4;           // row within the 128-row block
            const int pp  = i & 15;           // k-pair index
            const int row = coBase + r;
            const int k0  = kBase + 2 * pp;
            const int rc  = min(row, Co - 1);
            const float wx = Wb[(long)rc * wRow + (long)min(k0,     Ci - 1) * wCol];
            const float wy = Wb[(long)rc * wRow + (long)min(k0 + 1, Ci - 1) * wCol];
            v2f v;
            v.x = (row < Co && k0     < Ci) ? wx : 0.f;
            v.y = (row < Co && k0 + 1 < Ci) ? wy : 0.f;
            ws2[r][pp] = v;
        }
        __syncthreads();

        if (waveActive) {
            const int rowSel = wave * 16 + lane15;
            for (int k4 = 0; k4 < KP; k4 += 4) {
                const int pr = (k4 >> 1) + laneHalf;
                const v2f a  = ws2[rowSel][pr];          // one ds_load_b64
#pragma unroll
                for (int nt = 0; nt < 16; ++nt) {
                    const v2f bb = xs2[pr][nt * 16 + lane15];  // ds_load_b64
                    acc[nt] = __builtin_amdgcn_wmma_f32_16x16x4_f32(
                        false, a, false, bb, (short)0, acc[nt], false, false);
                }
            }
        }
        __syncthreads();
    }

    if (!waveActive) return;
    float* Yb = Y + (long)b * yBatch;
#pragma unroll
    for (int nt = 0; nt < 16; ++nt) {
#pragma unroll
        for (int r = 0; r < 8; ++r) {
            const int row = coBase + wave * 16 + r + 8 * laneHalf;  // D layout
            const int col = nt * 16 + lane15;
            if (row < Co) {
                float v = acc[nt][r] + (bias ? bias[row] : 0.f);
                if (bnG) v = bnG[row] * (v * BN_SCALE_C) + bnB[row];
                if (doRelu) v = fmaxf(v, 0.f);
                Yb[(long)row * yRowStride + col] = v;
            }
        }
    }
}

// ---------------------------------------------------------------------------
// center = xyz + pc(net)^T ; cls_prob = softmax over the 18 classes (in-place)
// one thread per (b,n)
// ---------------------------------------------------------------------------
__global__ __launch_bounds__(256) void softmax_center_kernel(
    const float* __restrict__ xyz, const float* __restrict__ pcO,
    float* __restrict__ psO, float* __restrict__ center)
{
    const int idx = blockIdx.x * 256 + threadIdx.x;
    if (idx >= BATCH * NPTS) return;
    const int b = idx / NPTS, n = idx % NPTS;

    const float* pcb = pcO + (long)b * 3 * NPTS;
#pragma unroll
    for (int d = 0; d < 3; ++d)
        center[(long)idx * 3 + d] = xyz[(long)idx * 3 + d] + pcb[d * NPTS + n];

    float* psb = psO + (long)b * NCLSC * NPTS;
    float mx = NEG_BIG;
    float e[NCLSC];
    for (int c = 0; c < NCLSC; ++c) mx = fmaxf(mx, psb[c * NPTS + n]);
    float s = 0.f;
    for (int c = 0; c < NCLSC; ++c) { e[c] = expf(psb[c * NPTS + n] - mx); s += e[c]; }
    const float inv = 1.f / s;
    for (int c = 0; c < NCLSC; ++c) psb[c * NPTS + n] = e[c] * inv;
}

// ---------------------------------------------------------------------------
// Top-16 neighbors per (b,n) (descending, ties -> lowest index, matching
// jax.lax.top_k), then edge geometry U = (d, atan2(dy,dx), atan2(dz,dx)).
// One workgroup per (b,n).
// ---------------------------------------------------------------------------
__global__ __launch_bounds__(256) void topk_edges_kernel(
    const float* __restrict__ sim, const float* __restrict__ center,
    int* __restrict__ tgt, float* __restrict__ U)
{
    __shared__ float vals[NPTS];
    __shared__ int   sel[KNBRC];

    const int b = blockIdx.x / NPTS;
    const int n = blockIdx.x % NPTS;
    const int m = threadIdx.x;

    // sim^T[n][m] == sim[m][n] (symmetric anyway)
    vals[m] = sim[((long)b * NPTS + m) * NPTS + n];
    __syncthreads();

    if (threadIdx.x == 0) {
        for (int k = 0; k < KNBRC; ++k) {
            float best = NEG_BIG; int bi = 0;
            for (int i = 0; i < NPTS; ++i) {
                float v = vals[i];
                if (v > best) { best = v; bi = i; }   // strict > : first occurrence
            }
            sel[k] = bi;
            vals[bi] = NEG_BIG;
        }
    }
    __syncthreads();

    if (threadIdx.x < KNBRC) {
        const int k = threadIdx.x;
        const int j = sel[k];
        const long e = (long)b * EDG + k * NPTS + n;   // tgt[b, k*N+n] = idx[b,n,k]
        tgt[e] = j;
        const float cix = center[((long)b * NPTS + n) * 3 + 0];
        const float ciy = center[((long)b * NPTS + n) * 3 + 1];
        const float ciz = center[((long)b * NPTS + n) * 3 + 2];
        const float dx = center[((long)b * NPTS + j) * 3 + 0] - cix;
        const float dy = center[((long)b * NPTS + j) * 3 + 1] - ciy;
        const float dz = center[((long)b * NPTS + j) * 3 + 2] - ciz;
        U[e * 3 + 0] = sqrtf(dx * dx + dy * dy + dz * dz);
        U[e * 3 + 1] = atan2f(dy, dx);
        U[e * 3 + 2] = atan2f(dz, dx);
    }
}

// ---------------------------------------------------------------------------
// GMM message pass, deterministic (no float atomics):
//   agg[n][c] = sum_k ( T_k @ G_k )[n][c],
//   T_k[n][j] = sum_{e -> n} wgt(e,k) * rep[src(e)][j]   (CSR, fixed order)
//   G_k[j][c] = gmm_g[j][k*C + c]
// T_k densified into global scratch; G_k staged in LDS (j-pair float2, 64KB);
// GEMM accumulated in WMMA regs across k.
// Finally fT[b][c][n] += agg[n][c] / max(indeg(n),1)  (fT holds root+bias).
// One workgroup per batch.
// ---------------------------------------------------------------------------
__global__ __launch_bounds__(256) void gmm_scatter_kernel(
    const int* __restrict__ tgt, const float* __restrict__ U,
    const float* __restrict__ repT, const float* __restrict__ gmm_g,
    const float* __restrict__ gmm_mu, const float* __restrict__ gmm_sigma,
    float* __restrict__ Tk, int* __restrict__ elist, float* __restrict__ fT)
{
    __shared__ v2f   Gp[CCH / 2][CCH];  // 64KB: Gp[jp][c] = {G[2jp][c], G[2jp+1][c]}
    __shared__ float wgtS[EDG];         // 16KB
    __shared__ int   offS[NPTS + 1];
    __shared__ int   cntS[NPTS];

    const int b   = blockIdx.x;
    const int tid = threadIdx.x;
    const int wave = tid >> 5, lane = tid & 31;
    const int laneHalf = lane >> 4, lane15 = lane & 15;

    const int*   tgtb = tgt + (long)b * EDG;
    const float* Ub   = U + (long)b * EDG * 3;
    const float* repB = repT + (long)b * CCH * NPTS;   // [j][n]
    float*       TkB  = Tk + (long)b * NPTS * CCH;
    int*         elB  = elist + (long)b * EDG;

    // deterministic CSR of incoming edges per target (thread tid == target n)
    {
        int c = 0;
        for (int e = 0; e < EDG; ++e) if (tgtb[e] == tid) ++c;
        cntS[tid] = c;
    }
    __syncthreads();
    if (tid == 0) {
        offS[0] = 0;
        for (int n = 0; n < NPTS; ++n) offS[n + 1] = offS[n] + cntS[n];
    }
    __syncthreads();
    {
        int w = offS[tid];
        for (int e = 0; e < EDG; ++e) if (tgtb[e] == tid) elB[w++] = e;
    }
    __threadfence();
    __syncthreads();

    v8f acc[16];
    const v8f vzero = {0.f,0.f,0.f,0.f,0.f,0.f,0.f,0.f};
#pragma unroll
    for (int t = 0; t < 16; ++t) acc[t] = vzero;

    for (int k = 0; k < KGC_K; ++k) {
        const float mu0 = gmm_mu[k*3+0], mu1 = gmm_mu[k*3+1], mu2 = gmm_mu[k*3+2];
        const float s0 = gmm_sigma[k*3+0], s1 = gmm_sigma[k*3+1], s2 = gmm_sigma[k*3+2];
        const float iv0 = 1.f/(s0*s0+GEPS_F), iv1 = 1.f/(s1*s1+GEPS_F), iv2 = 1.f/(s2*s2+GEPS_F);

        // edge weights for component k
        for (int e = tid; e < EDG; e += 256) {
            const float d0 = Ub[e*3+0] - mu0;
            const float d1 = Ub[e*3+1] - mu1;
            const float d2 = Ub[e*3+2] - mu2;
            wgtS[e] = expf(-0.5f * (d0*d0*iv0 + d1*d1*iv1 + d2*d2*iv2));
        }
        __syncthreads();

        // densify T_k (full overwrite -> no zeroing needed, deterministic order)
        for (int p = tid; p < NPTS * CCH; p += 256) {
            const int n = p >> 7;
            const int j = p & 127;
            float s = 0.f;
            const int e0 = offS[n], e1 = offS[n + 1];
            for (int t = e0; t < e1; ++t) {
                const int e = elB[t];
                s += wgtS[e] * repB[j * NPTS + (e & 255)];   // src(e) = e % N
            }
            TkB[n * CCH + j] = s;
        }
        // stage G_k into LDS, j-pair packed (coalesced along c across lanes)
        for (int i = tid; i < (CCH / 2) * CCH; i += 256) {
            const int jp = i >> 7;     // 0..63
            const int c  = i & 127;
            const long base = (long)(2 * jp) * KGC + (long)k * CCH + c;
            v2f v;
            v.x = gmm_g[base];
            v.y = gmm_g[base + KGC];
            Gp[jp][c] = v;
        }
        __threadfence();
        __syncthreads();

        // acc += T_k @ G_k  (M=256 targets, N=128 channels, K=128)
        const int rowBase = wave * 32 + lane15;     // wave owns M-tiles 2w, 2w+1
        for (int j0 = 0; j0 < CCH; j0 += 4) {
            const int jA = j0 + 2 * laneHalf;       // even -> 8B aligned pairs
            const int jp = (j0 >> 1) + laneHalf;
            const v2f a0 = *reinterpret_cast<const v2f*>(TkB + rowBase * CCH + jA);
            const v2f a1 = *reinterpret_cast<const v2f*>(TkB + (rowBase + 16) * CCH + jA);
#pragma unroll
            for (int ct = 0; ct < 8; ++ct) {
                const v2f bb = Gp[jp][ct * 16 + lane15];     // one ds_load_b64
                acc[ct]     = __builtin_amdgcn_wmma_f32_16x16x4_f32(
                    false, a0, false, bb, (short)0, acc[ct], false, false);
                acc[8 + ct] = __builtin_amdgcn_wmma_f32_16x16x4_f32(
                    false, a1, false, bb, (short)0, acc[8 + ct], false, false);
            }
        }
        __syncthreads();
    }

    // fT[b][c][n] += agg[n][c] / max(indeg(n),1)
    float* fTb = fT + (long)b * CCH * NPTS;
#pragma unroll
    for (int half = 0; half < 2; ++half) {
        const int nBase = wave * 32 + half * 16;
#pragma unroll
        for (int ct = 0; ct < 8; ++ct) {
#pragma unroll
            for (int r = 0; r < 8; ++r) {
                const int n = nBase + r + 8 * laneHalf;
                const int c = ct * 16 + lane15;
                const float cn = (float)(offS[n + 1] - offS[n]);
                const float rinv = 1.f / fmaxf(cn, 1.f);
                const long o = (long)c * NPTS + n;
                fTb[o] = fTb[o] + acc[half * 8 + ct][r] * rinv;
            }
        }
    }
}

// ---------------------------------------------------------------------------
// Slice conv3 output into the 9 concatenated outputs.
// ---------------------------------------------------------------------------
__global__ __launch_bounds__(256) void assemble_kernel(
    const float* __restrict__ nt, const float* __restrict__ xyz,
    const float* __restrict__ msz, float* __restrict__ out)
{
    const int idx = blockIdx.x * 256 + threadIdx.x;
    if (idx >= BATCH * NPTS) return;
    const int b = idx / NPTS, n = idx % NPTS;
    const float* ntb = nt + (long)b * COUT3 * NPTS;

    float* obj  = out + 0;
    float* cen  = out + 8192;
    float* hs   = out + 20480;
    float* hrn  = out + 24576;
    float* hr   = out + 28672;
    float* ss   = out + 32768;
    float* srn  = out + 106496;
    float* sr   = out + 327680;
    float* sem  = out + 548864;

    obj[(long)idx*2+0] = ntb[0*NPTS+n];
    obj[(long)idx*2+1] = ntb[1*NPTS+n];
#pragma unroll
    for (int d = 0; d < 3; ++d)
        cen[(long)idx*3+d] = xyz[(long)idx*3+d] + ntb[(2+d)*NPTS+n];
    hs[idx]  = ntb[5*NPTS+n];
    const float hv = ntb[6*NPTS+n];
    hrn[idx] = hv;
    hr[idx]  = hv * PI_F;                 // pi / NHB, NHB == 1
    for (int i = 0; i < NSCC; ++i)
        ss[(long)idx*NSCC+i] = ntb[(7+i)*NPTS+n];
    for (int i = 0; i < 3*NSCC; ++i) {
        const float v = ntb[(25+i)*NPTS+n];
        srn[(long)idx*3*NSCC+i] = v;
        sr [(long)idx*3*NSCC+i] = v * msz[i];
    }
    for (int i = 0; i < NSCC; ++i)
        sem[(long)idx*NSCC+i] = ntb[(79+i)*NPTS+n];
}

// ---------------------------------------------------------------------------
// Host launcher
// ---------------------------------------------------------------------------
extern "C" void kernel_launch(void* const* d_in, const int* in_sizes, int n_in,
                              void* d_out, int out_size, void* d_ws, size_t ws_size,
                              hipStream_t stream) {
    (void)in_sizes; (void)n_in; (void)out_size; (void)ws_size;
    const float* xyz     = (const float*)d_in[0];
    const float* feat    = (const float*)d_in[1];
    const float* conv1_w = (const float*)d_in[2];
    const float* conv1_b = (const float*)d_in[3];
    const float* bn1_g   = (const float*)d_in[4];
    const float* bn1_b   = (const float*)d_in[5];
    const float* conv2_w = (const float*)d_in[6];
    const float* conv2_b = (const float*)d_in[7];
    const float* bn2_g   = (const float*)d_in[8];
    const float* bn2_b   = (const float*)d_in[9];
    const float* pc_w    = (const float*)d_in[10];
    const float* pc_b    = (const float*)d_in[11];
    const float* ps_w    = (const float*)d_in[12];
    const float* ps_b    = (const float*)d_in[13];
    const float* rf1_w   = (const float*)d_in[14];
    const float* rf1_b   = (const float*)d_in[15];
    const float* rf2_w   = (const float*)d_in[16];
    const float* rf2_b   = (const float*)d_in[17];
    const float* gmm_g   = (const float*)d_in[18];
    const float* gmm_mu  = (const float*)d_in[19];
    const float* gmm_sig = (const float*)d_in[20];
    const float* gmm_root= (const float*)d_in[21];
    const float* gmm_bias= (const float*)d_in[22];
    const float* sg1_w   = (const float*)d_in[23];
    const float* sg1_b   = (const float*)d_in[24];
    const float* sg2_w   = (const float*)d_in[25];
    const float* sg2_b   = (const float*)d_in[26];
    const float* c35_w   = (const float*)d_in[27];
    const float* c35_b   = (const float*)d_in[28];
    const float* bn3_g   = (const float*)d_in[29];
    const float* bn3_b   = (const float*)d_in[30];
    const float* conv3_w = (const float*)d_in[31];
    const float* conv3_b = (const float*)d_in[32];
    const float* msz     = (const float*)d_in[33];

    float* ws  = (float*)d_ws;
    float* out = (float*)d_out;

    // ws layout (floats), with lifetime-based aliasing:
    float* net1   = ws + 0;        // 524288 : net1, later represent^T
    float* cat    = ws + 524288;   // 1048576: [net2 ; h] concat (B,256,N)
    float* pcO    = ws + 1572864;  // 12288
    float* psO    = ws + 1585152;  // 73728  : ps logits -> cls_prob (in place)
    float* r1     = ws + 1658880;  // 1048576: rf1-out, then sim, then sg1-out
    float* zc     = ws + 2707456;  // 524288 : z^T, later fT
    float* cenB   = ws + 3231744;  // 12288
    int*   tgt    = (int*)(ws + 3244032); // 65536 ints
    float* Ub     = ws + 3309568;  // 196608
    float* nnB    = ws + 3506176;  // 524288: T_k scratch, later new_net
    float* ntB    = ws + 4030464;  // 397312
    int*   elist  = (int*)(ws + 4427776); // 65536 ints   (total ~18 MB)

    const long CN  = (long)CCH * NPTS;     // 32768
    const long C2N = 2 * CN;               // 65536
    dim3 blk(256);

    // net = relu(bn1(conv1(features)))
    gemm_ws<<<dim3(BATCH,1),blk,0,stream>>>(conv1_w,0,CCH,1, feat,CN,NPTS,1,
        conv1_b,bn1_g,bn1_b, net1,CN,NPTS, CCH,CCH,1);
    // net = relu(bn2(conv2(net)))  -> rows 0..127 of cat
    gemm_ws<<<dim3(BATCH,1),blk,0,stream>>>(conv2_w,0,CCH,1, net1,CN,NPTS,1,
        conv2_b,bn2_g,bn2_b, cat,C2N,NPTS, CCH,CCH,1);
    // pc / ps heads
    gemm_ws<<<dim3(BATCH,1),blk,0,stream>>>(pc_w,0,CCH,1, cat,C2N,NPTS,1,
        pc_b,nullptr,nullptr, pcO,(long)3*NPTS,NPTS, 3,CCH,0);
    gemm_ws<<<dim3(BATCH,1),blk,0,stream>>>(ps_w,0,CCH,1, cat,C2N,NPTS,1,
        ps_b,nullptr,nullptr, psO,(long)NCLSC*NPTS,NPTS, NCLSC,CCH,0);
    // z^T = relu(rf2(rf1(net)))
    gemm_ws<<<dim3(BATCH,1),blk,0,stream>>>(rf1_w,0,CCH,1, cat,C2N,NPTS,1,
        rf1_b,nullptr,nullptr, r1,CN,NPTS, CCH,CCH,0);
    gemm_ws<<<dim3(BATCH,1),blk,0,stream>>>(rf2_w,0,CCH,1, r1,CN,NPTS,1,
        rf2_b,nullptr,nullptr, zc,CN,NPTS, CCH,CCH,1);
    // center + softmax
    softmax_center_kernel<<<dim3(BATCH),blk,0,stream>>>(xyz,pcO,psO,cenB);
    // sim = z z^T  (batched-W WMMA GEMM)
    gemm_ws<<<dim3(BATCH,2),blk,0,stream>>>(zc,CN,1,NPTS, zc,CN,NPTS,1,
        nullptr,nullptr,nullptr, r1,(long)NPTS*NPTS,NPTS, NPTS,CCH,0);
    // top-k neighbors + edge geometry
    topk_edges_kernel<<<dim3(BATCH*NPTS),blk,0,stream>>>(r1,cenB,tgt,Ub);
    // represent^T = (cls_prob @ ps_w)^T : W[o=d][i=k] = ps_w[k*C+d]
    gemm_ws<<<dim3(BATCH,1),blk,0,stream>>>(ps_w,0,1,CCH, psO,(long)NCLSC*NPTS,NPTS,1,
        nullptr,nullptr,nullptr, net1,CN,NPTS, CCH,NCLSC,0);
    // fT = (represent @ gmm_root)^T + gmm_bias : W[o=c][i=j] = gmm_root[j*C+c]
    gemm_ws<<<dim3(BATCH,1),blk,0,stream>>>(gmm_root,0,1,CCH, net1,CN,NPTS,1,
        gmm_bias,nullptr,nullptr, zc,CN,NPTS, CCH,CCH,0);
    // fT += agg / max(cnt,1)
    gmm_scatter_kernel<<<dim3(BATCH),blk,0,stream>>>(tgt,Ub,net1,gmm_g,
        gmm_mu,gmm_sig, nnB, elist, zc);
    // h = relu(sg2(relu(sg1(fT))))  -> rows 128..255 of cat
    gemm_ws<<<dim3(BATCH,2),blk,0,stream>>>(sg1_w,0,CCH,1, zc,CN,NPTS,1,
        sg1_b,nullptr,nullptr, r1,C2N,NPTS, 2*CCH,CCH,1);
    gemm_ws<<<dim3(BATCH,1),blk,0,stream>>>(sg2_w,0,2*CCH,1, r1,C2N,NPTS,1,
        sg2_b,nullptr,nullptr, cat + (long)CCH*NPTS,C2N,NPTS, CCH,2*CCH,1);
    // new_net = relu(bn3(c35(cat)))
    gemm_ws<<<dim3(BATCH,1),blk,0,stream>>>(c35_w,0,2*CCH,1, cat,C2N,NPTS,1,
        c35_b,bn3_g,bn3_b, nnB,CN,NPTS, CCH,2*CCH,1);
    // nt = conv3(new_net)
    gemm_ws<<<dim3(BATCH,1),blk,0,stream>>>(conv3_w,0,CCH,1, nnB,CN,NPTS,1,
        conv3_b,nullptr,nullptr, ntB,(long)COUT3*NPTS,NPTS, COUT3,CCH,0);
    // slice into the 9 outputs
    assemble_kernel<<<dim3(BATCH),blk,0,stream>>>(ntB,xyz,msz,out);
}